// SAGEBlock_43920335568925
// MI455X (gfx1250) — compile-verified
//
#include <hip/hip_runtime.h>

#define N_NODES 50000
#define D_FEAT  128
#define BN_EPS  1e-5f

// K-pairs per 128-deep GEMM, padded row stride (in float2) for bank-conflict-free b64 LDS loads
#define NPAIR   64
#define PSTRIDE 144   // 128 rows + 16 pad; 288 dwords => +32 banks between adjacent pairs

typedef __attribute__((ext_vector_type(2))) float v2f;
typedef __attribute__((ext_vector_type(8))) float v8f;

// ---------------------------------------------------------------------------
// Kernel 1: zero the accumulators in workspace (agg, deg, colsum, colsq)
// ---------------------------------------------------------------------------
__global__ void k_zero(float* __restrict__ p, int n) {
    int i = blockIdx.x * blockDim.x + threadIdx.x;
    if (i < n) p[i] = 0.0f;
}

// ---------------------------------------------------------------------------
// Kernel 2: edge scatter-add. One wave32 per edge: each lane gathers a float4
// of x[src] and does 4 native f32 atomic adds into agg[dst]. x and agg are
// L2-resident (25.6 MB each vs 192 MB L2) -> L2-atomic-throughput bound.
// ---------------------------------------------------------------------------
__global__ void k_scatter(const float4* __restrict__ x4,
                          const int*    __restrict__ src,
                          const int*    __restrict__ dst,
                          float*        __restrict__ agg,
                          float*        __restrict__ deg,
                          int E) {
    int gw   = (blockIdx.x * blockDim.x + threadIdx.x) >> 5;  // wave id = edge id
    int lane = threadIdx.x & 31;
    if (gw >= E) return;
    int s = src[gw];
    int d = dst[gw];
    float4 v = x4[(size_t)s * (D_FEAT / 4) + lane];
    float* a = agg + (size_t)d * D_FEAT + lane * 4;
    unsafeAtomicAdd(a + 0, v.x);
    unsafeAtomicAdd(a + 1, v.y);
    unsafeAtomicAdd(a + 2, v.z);
    unsafeAtomicAdd(a + 3, v.w);
    if (lane == 0) unsafeAtomicAdd(deg + d, 1.0f);
}

// ---------------------------------------------------------------------------
// Kernel 3: h = (agg + x)/(deg+1);  pre = x + h @ W^T + b;  column sum/sumsq.
// 256 threads = 8 waves, 128 rows per block. LDS holds K-pair-interleaved
// W^T (async-copied from global with GLOBAL_LOAD_ASYNC_TO_LDS_B64) and the
// h tile. Fragments are double-buffered in VGPRs: step kt+1's A/B fragments
// are ds_load_b64'd behind step kt's 8-WMMA burst, so DS latency is covered
// once per k-step instead of gating every WMMA pair.
// ---------------------------------------------------------------------------
__global__ __launch_bounds__(256)
void k_sage_gemm(const float* __restrict__ x,
                 const float* __restrict__ W,
                 const float* __restrict__ bvec,
                 const float* __restrict__ agg,
                 const float* __restrict__ deg,
                 float*       __restrict__ outp,
                 float*       __restrict__ colsum,
                 float*       __restrict__ colsq) {
    extern __shared__ v2f smem2[];
    v2f* Wt2 = smem2;                    // Wt2[p*PSTRIDE + j] = {W[j][2p], W[j][2p+1]}
    v2f* ht2 = smem2 + NPAIR * PSTRIDE;  // ht2[p*PSTRIDE + r] = {h[r][2p], h[r][2p+1]}

    const int tid  = threadIdx.x;
    const int row0 = blockIdx.x * 128;

    // --- Async-stage W^T into LDS (pair-swizzled scatter), tracked by ASYNCcnt.
    // Low 32 bits of a flat LDS pointer are the LDS byte offset (ISA 10.2).
    for (int idx = tid; idx < D_FEAT * NPAIR; idx += 256) {
        int j = idx >> 6, p = idx & (NPAIR - 1);           // idx = j*64 + p (coalesced global)
        unsigned ldsoff = (unsigned)(uintptr_t)(Wt2 + p * PSTRIDE + j);
        unsigned goff   = (unsigned)idx * 8u;              // byte offset of W[j][2p]
        asm volatile("global_load_async_to_lds_b64 %0, %1, %2"
                     :: "v"(ldsoff), "v"(goff), "s"(W)
                     : "memory");
    }

    // --- Overlap: compute normalized features into LDS (needs VALU anyway).
    for (int idx = tid; idx < 128 * NPAIR; idx += 256) {
        int r = idx >> 6, p = idx & (NPAIR - 1);
        int row = row0 + r;
        v2f h = {0.0f, 0.0f};
        if (row < N_NODES) {
            float inv = 1.0f / (deg[row] + 1.0f);
            v2f av = ((const v2f*)agg)[(size_t)row * NPAIR + p];
            v2f xv = ((const v2f*)x)[(size_t)row * NPAIR + p];
            h.x = (av.x + xv.x) * inv;
            h.y = (av.y + xv.y) * inv;
        }
        ht2[p * PSTRIDE + r] = h;
    }

    asm volatile("s_wait_asynccnt 0x0" ::: "memory");
    __syncthreads();

    const int wave  = tid >> 5;
    const int lane  = tid & 31;
    const int rsel  = lane & 15;        // row (A) / col (B) within 16-tile
    const int phalf = lane >> 4;        // lanes 16-31 take the next K-pair
    const int rbase = wave << 4;        // this wave's 16-row slab

    v8f acc[8];
#pragma unroll
    for (int i = 0; i < 8; ++i) acc[i] = (v8f){0, 0, 0, 0, 0, 0, 0, 0};

    int ai = phalf * PSTRIDE + rbase + rsel;   // A fragment index (v2f units)
    int bi = phalf * PSTRIDE + rsel;           // B fragment base index

    // Preload fragments for kt = 0.
    v2f a = ht2[ai];
    v2f bf[8];
#pragma unroll
    for (int ct = 0; ct < 8; ++ct) bf[ct] = Wt2[bi + (ct << 4)];

#pragma unroll
    for (int kt = 0; kt < 31; ++kt) {
        ai += 2 * PSTRIDE;
        bi += 2 * PSTRIDE;
        // Prefetch fragments for kt+1 (hidden behind this step's WMMAs).
        v2f an = ht2[ai];
        v2f bn[8];
#pragma unroll
        for (int ct = 0; ct < 8; ++ct) bn[ct] = Wt2[bi + (ct << 4)];
#pragma unroll
        for (int ct = 0; ct < 8; ++ct)
            acc[ct] = __builtin_amdgcn_wmma_f32_16x16x4_f32(
                false, a, false, bf[ct], (short)0, acc[ct], false, false);
        a = an;
#pragma unroll
        for (int ct = 0; ct < 8; ++ct) bf[ct] = bn[ct];
    }
    // Final k-step.
#pragma unroll
    for (int ct = 0; ct < 8; ++ct)
        acc[ct] = __builtin_amdgcn_wmma_f32_16x16x4_f32(
            false, a, false, bf[ct], (short)0, acc[ct], false, false);

    // Epilogue: residual + bias, store pre-BN output, column sum/sum^2.
    // C/D layout: VGPR r -> row M=r (lanes 0-15) / M=r+8 (lanes 16-31), N=lane%16.
    const int rhalf = phalf << 3;
#pragma unroll
    for (int ct = 0; ct < 8; ++ct) {
        const int col = (ct << 4) + rsel;
        const float bb = bvec[col];
        float lsum = 0.0f, lsq = 0.0f;
#pragma unroll
        for (int r = 0; r < 8; ++r) {
            int row = row0 + rbase + r + rhalf;
            if (row < N_NODES) {
                float val = acc[ct][r] + x[(size_t)row * D_FEAT + col] + bb;
                outp[(size_t)row * D_FEAT + col] = val;
                lsum += val;
                lsq  += val * val;
            }
        }
        lsum += __shfl_xor(lsum, 16, 32);   // fold lanes 16-31 onto 0-15 (same col)
        lsq  += __shfl_xor(lsq, 16, 32);
        if (lane < 16) {
            unsafeAtomicAdd(&colsum[col], lsum);
            unsafeAtomicAdd(&colsq[col], lsq);
        }
    }
}

// ---------------------------------------------------------------------------
// Kernel 4: turn column sums into (scale, shift) for BN. 128 threads.
// ---------------------------------------------------------------------------
__global__ void k_bn_finalize(float* __restrict__ colsum, float* __restrict__ colsq,
                              const float* __restrict__ gamma,
                              const float* __restrict__ beta) {
    int c = threadIdx.x;
    if (c < D_FEAT) {
        const float invN = 1.0f / (float)N_NODES;
        float mean  = colsum[c] * invN;
        float var   = colsq[c] * invN - mean * mean;
        float scale = gamma[c] * rsqrtf(var + BN_EPS);
        float shift = beta[c] - mean * scale;
        colsum[c] = scale;   // reuse slots
        colsq[c]  = shift;
    }
}

// ---------------------------------------------------------------------------
// Kernel 5: y = relu(pre * scale[c] + shift[c]) streaming over N*D.
// ---------------------------------------------------------------------------
__global__ void k_bn_relu(float* __restrict__ outp,
                          const float* __restrict__ scale,
                          const float* __restrict__ shift,
                          int n) {
    int i = blockIdx.x * blockDim.x + threadIdx.x;
    if (i < n) {
        int c = i & (D_FEAT - 1);
        float v = fmaf(outp[i], scale[c], shift[c]);
        outp[i] = v > 0.0f ? v : 0.0f;
    }
}

// ---------------------------------------------------------------------------
extern "C" void kernel_launch(void* const* d_in, const int* in_sizes, int n_in,
                              void* d_out, int out_size, void* d_ws, size_t ws_size,
                              hipStream_t stream) {
    const float* x     = (const float*)d_in[0];
    const int*   src   = (const int*)  d_in[1];
    const int*   dst   = (const int*)  d_in[2];
    const float* W     = (const float*)d_in[3];
    const float* bvec  = (const float*)d_in[4];
    const float* gamma = (const float*)d_in[5];
    const float* beta  = (const float*)d_in[6];
    float*       outp  = (float*)d_out;
    const int    E     = in_sizes[1];

    // Workspace layout (floats): agg[N*D] | deg[N] | colsum[128] | colsq[128]
    float* agg    = (float*)d_ws;
    float* deg    = agg + (size_t)N_NODES * D_FEAT;
    float* colsum = deg + N_NODES;
    float* colsq  = colsum + D_FEAT;

    const int nzero = N_NODES * D_FEAT + N_NODES + 2 * D_FEAT;
    k_zero<<<(nzero + 255) / 256, 256, 0, stream>>>(agg, nzero);

    const int scatter_blocks = (E + 7) / 8;  // 8 edges (waves) per 256-thread block
    k_scatter<<<scatter_blocks, 256, 0, stream>>>(
        (const float4*)x, src, dst, agg, deg, E);

    const int gemm_blocks = (N_NODES + 127) / 128;
    const size_t lds_bytes = 2u * NPAIR * PSTRIDE * sizeof(v2f);  // 144 KB
    k_sage_gemm<<<gemm_blocks, 256, lds_bytes, stream>>>(
        x, W, bvec, agg, deg, outp, colsum, colsq);

    k_bn_finalize<<<1, D_FEAT, 0, stream>>>(colsum, colsq, gamma, beta);

    const int n = N_NODES * D_FEAT;
    k_bn_relu<<<(n + 255) / 256, 256, 0, stream>>>(outp, colsum, colsq, n);
}